// CAM_64415919505626
// MI455X (gfx1250) — compile-verified
//
#include <hip/hip_runtime.h>

// CAM perception kernel for MI455X (gfx1250, wave32, WMMA).
// out[b, c, n] = gamma * sum_d ( rowmax_d'(P[b,c,d']) - P[b,c,d] ) * img[b,d,n] + img[b,c,n]
// where P[b,c,d] = sum_n txt[b,c,n] * txt[b,d,n],  C=8, N=2048, D=C*N=16384.
//
// One 256-thread block (8 wave32s) per batch row. Memory-bound (768 MB total,
// ~33us floor at 23.3 TB/s). All matrix math via V_WMMA_F32_16X16X4_F32.
//
// Zero-padding of the 8-row matrices into the 16-row WMMA tiles is replaced by
// ROW DUPLICATION (row = lane & 7): D[m][n] = P[m&7][n&7], the 8x8 corner is
// unchanged, loads become branchless (no exec diamonds), and duplicate lanes
// hit identical addresses (coalesced away).

typedef __attribute__((ext_vector_type(2))) float v2f;
typedef __attribute__((ext_vector_type(8))) float v8f;

#define CAM_C   8
#define CAM_N   2048      // per-channel length
#define CAM_D   16384     // C*N

__global__ __launch_bounds__(256) void
CAM_64415919505626_kernel(const float* __restrict__ img,
                          const float* __restrict__ txt,
                          const float* __restrict__ gamma,
                          float* __restrict__ out)
{
    __shared__ float sRed[8 * 64];  // per-wave partial 8x8 gram blocks
    __shared__ float sP[64];        // reduced gram P[c][d]
    __shared__ float sM[64];        // fused epilogue matrix M[c][d]

    const int tid  = threadIdx.x;
    const int wave = tid >> 5;      // 8 waves of 32 (gfx1250 is wave32)
    const int lane = tid & 31;
    const int half = lane >> 4;     // 0: lanes 0-15 (K0/K1), 1: lanes 16-31 (K2/K3)
    const int l16  = lane & 15;     // tile row (A) / tile column (B, D)

    const size_t base = (size_t)blockIdx.x * CAM_D;
    const float* q  = txt + base;   // Q: [8][2048]
    const float* im = img + base;   // V: [8][2048]
    float*       ob = out + base;

    // ---------------- Phase 1: P = Q * Q^T via WMMA f32 16x16x4 ----------------
    // A-tile 16x4 fp32 layout: lane l<16 holds A[M=l][K=0..1] in {x,y};
    // lane 16+l holds A[M=l][K=2..3]. For a Gram matrix the B-tile (4x16)
    // register image is identical to A, so one v2f load per lane serves both.
    // Tile rows 8..15 duplicate rows 0..7 (same addresses -> coalesced).
    const float* qrow = q + (l16 & 7) * CAM_N + 2 * half;

    v8f acc = {};                           // 16x16 f32 accumulator (8 VGPRs)
    const int k0 = wave * (CAM_N / 8);      // each wave owns a K-slice of 256
    #pragma unroll 4
    for (int k = k0; k < k0 + CAM_N / 8; k += 4) {
        v2f a = *(const v2f*)(qrow + k);    // 8B aligned, branchless
        acc = __builtin_amdgcn_wmma_f32_16x16x4_f32(
                  false, a, false, a, (short)0, acc, false, false);
    }

    // Stash the useful 8x8 corner (M=0..7 -> VGPRs 0..7, N=lane 0..7).
    if (half == 0 && l16 < CAM_C) {
        #pragma unroll
        for (int r = 0; r < CAM_C; ++r)
            sRed[wave * 64 + r * 8 + l16] = acc[r];
    }
    __syncthreads();

    // ---------------- Phase 2: reduce waves, rowmax, fold gamma + identity ----
    if (tid < 64) {
        float p = 0.f;
        #pragma unroll
        for (int w = 0; w < 8; ++w) p += sRed[w * 64 + tid];
        sP[tid] = p;
    }
    __syncthreads();
    if (tid < 64) {
        const int c = tid >> 3, d = tid & 7;
        const float* rowp = &sP[c * 8];
        float m = rowp[0];
        #pragma unroll
        for (int j = 1; j < 8; ++j) m = fmaxf(m, rowp[j]);
        const float g = gamma[0];
        // M = gamma*(rowmax - P) + I  =>  out = M @ img (residual folded in)
        sM[tid] = g * (m - sP[tid]) + ((c == d) ? 1.0f : 0.0f);
    }
    __syncthreads();

    // ---------------- Phase 3: out = M @ img via WMMA (A loop-invariant) ------
    // A rows 8..15 duplicate rows 0..7; duplicated D rows are never stored.
    const int rM = (l16 & 7) * 8;
    v2f a0, a1;
    a0.x = sM[rM + 2 * half + 0];           // chunk0: K = d 0..3
    a0.y = sM[rM + 2 * half + 1];
    a1.x = sM[rM + 2 * half + 4];           // chunk1: K = d 4..7
    a1.y = sM[rM + 2 * half + 5];

    // 128 column tiles of 16; wave-strided. B-tile 4x16: VGPR0 lanes0-15 = row K0,
    // lanes16-31 = row K2; VGPR1 = K1/K3 -> 4 coalesced 64B loads per 16 lanes.
    #pragma unroll 2
    for (int t = wave; t < CAM_N / 16; t += 8) {
        const int n0 = t * 16;
        const float* bp = im + n0 + l16;
        v2f b0, b1;
        b0.x = bp[(0 + 2 * half) * CAM_N];
        b0.y = bp[(1 + 2 * half) * CAM_N];
        b1.x = bp[(4 + 2 * half) * CAM_N];
        b1.y = bp[(5 + 2 * half) * CAM_N];

        v8f d = {};
        d = __builtin_amdgcn_wmma_f32_16x16x4_f32(
                false, a0, false, b0, (short)0, d, false, false);
        d = __builtin_amdgcn_wmma_f32_16x16x4_f32(
                false, a1, false, b1, (short)0, d, false, false);

        // D rows 0..7 live in VGPRs 0..7 of lanes 0..15 (rows 8..15 are dups).
        if (half == 0) {
            float* op = ob + n0 + l16;
            #pragma unroll
            for (int r = 0; r < CAM_C; ++r)
                __builtin_nontemporal_store(d[r], op + r * CAM_N);
        }
    }
}

extern "C" void kernel_launch(void* const* d_in, const int* in_sizes, int n_in,
                              void* d_out, int out_size, void* d_ws, size_t ws_size,
                              hipStream_t stream) {
    const float* img   = (const float*)d_in[0];
    const float* txt   = (const float*)d_in[1];
    const float* gamma = (const float*)d_in[2];
    float* out = (float*)d_out;

    const int B = in_sizes[0] / CAM_D;   // 4096
    dim3 grid(B), block(256);
    hipLaunchKernelGGL(CAM_64415919505626_kernel, grid, block, 0, stream,
                       img, txt, gamma, out);
}